// Conv3D_54254026883672
// MI455X (gfx1250) — compile-verified
//
#include <hip/hip_runtime.h>

typedef __attribute__((ext_vector_type(2))) float v2f;
typedef __attribute__((ext_vector_type(8))) float v8f;

#define CIN 3
#define KDN 5
#define KHN 7
#define KWN 7
#define KWP 8                       // kw padded 7 -> 8 (pad tap contributes 0)
#define KTAPS (CIN * KDN * KHN * KWP)   // 840
#define DIN 16
#define HIN 112
#define WIN 112
#define DOUT 16
#define HOUT 56
#define WOUT 56
#define NBATCH 8
#define COUT 64
#define SPATIAL (HOUT * WOUT)       // 3136
#define PTILES (SPATIAL / 16)       // 196 (exact)

#define CHUNK_PAIRS 84              // 5 chunks * 84 = 420 pairs
#define CHUNK_ITERS 42              // WMMA steps per chunk (2 pairs each)
#define NCHUNKS 5
#define LDS_FLOATS (CHUNK_PAIRS * 128)  // 10752 floats = 43008 B (< 64KB)

// Pack weight (64,3,5,7,7) into WMMA-A-friendly layout:
//   wpk[ (pair*64 + co)*2 + (t&1) ],  pair = t>>1,  t in [0,840)
// where t = ((ci*5 + kd)*7 + kh)*8 + kw, and kw==7 is a zero pad tap.
__global__ void pack_weights(const float* __restrict__ w, float* __restrict__ wpk) {
    int idx = blockIdx.x * blockDim.x + threadIdx.x;
    if (idx >= KTAPS * COUT) return;
    int e    = idx & 1;
    int tmp  = idx >> 1;
    int co   = tmp & 63;
    int pair = tmp >> 6;
    int t    = pair * 2 + e;
    int kw   = t & 7;
    int r    = t >> 3;
    int kh   = r % 7;
    int r2   = r / 7;
    int kd   = r2 % 5;
    int ci   = r2 / 5;
    float val = 0.0f;
    if (kw < 7)
        val = w[co * (CIN * KDN * KHN * KWN) + ((ci * KDN + kd) * KHN + kh) * KWN + kw];
    wpk[idx] = val;
}

// Branchless predicated gather: clamp address, always load, cndmask result.
__device__ __forceinline__ float gather(const float* __restrict__ x,
                                        int addr, bool valid) {
    float v = x[valid ? addr : 0];
    return valid ? v : 0.0f;
}

// One wave computes a 64(M) x 16(N) output tile:
//   M = output channels (4 WMMA tiles), N = 16 flattened (h,w) positions,
//   K = 840 padded taps -> 210 steps of V_WMMA_F32_16X16X4_F32.
// Single WMMA sub-step per iteration (avoids accumulator bank copies),
// fully branchless inner loop.
__global__ __launch_bounds__(256) void conv3d_wmma(
        const float* __restrict__ x, const float* __restrict__ wpk,
        const float* __restrict__ bias, float* __restrict__ out) {
    __shared__ float wlds[LDS_FLOATS];

    const int tid  = threadIdx.x;
    const int lane = tid & 31;
    const int lo   = lane & 15;   // N index / M row within tile
    const int hi   = lane >> 4;   // selects K pair (0/1) and M half for C
    const int wv   = tid >> 5;

    const int gid   = blockIdx.x * 8 + wv;      // 25088 wave tiles total
    const int ptile = gid % PTILES;
    const int rest  = gid / PTILES;
    const int d     = rest & 15;
    const int n     = rest >> 4;

    const int p  = ptile * 16 + lo;
    const int h  = p / WOUT;
    const int w  = p - h * WOUT;
    const int ih0     = 2 * h - 3;              // + kh
    const int iw0base = 2 * w - 3 + 2 * hi;     // + kwBase -> kw0 element

    // ---- loop-invariant per-lane window masks (kwBase = 0 / 4) ----
    const bool vw0A = ((unsigned)iw0base       < WIN);
    const bool vw1A = ((unsigned)(iw0base + 1) < WIN);
    const bool vw0B = ((unsigned)(iw0base + 4) < WIN);
    const bool vw1B = ((unsigned)(iw0base + 5) < WIN) & (hi == 0); // kw==7 pad

    // per-lane base address: xn + (d-2)*HW + ih0*W + iw0base
    const int laneBase = n * (CIN * DIN * HIN * WIN)
                       + (d - 2) * (HIN * WIN)
                       + ih0 * WIN + iw0base;

    const v2f* wlds2 = (const v2f*)wlds;

    v8f acc0 = {}, acc1 = {}, acc2 = {}, acc3 = {};

    int kh = 0, kd = 0, ci = 0;     // uniform tap state (SALU)

    for (int c = 0; c < NCHUNKS; ++c) {
        __syncthreads();
        {   // cooperative stage of 84 K-pairs of packed weights into LDS
            const float* src = wpk + c * (CHUNK_PAIRS * 128);
            #pragma unroll
            for (int j = 0; j < LDS_FLOATS / 256; ++j)
                wlds[tid + j * 256] = src[tid + j * 256];
        }
        __syncthreads();

        for (int il = 0; il < CHUNK_ITERS; ++il) {
            const int par    = il & 1;           // uniform parity
            const int kwBase = par << 2;         // 0 or 4

            // ---- uniform (SALU) part of the tap address / validity ----
            const bool vd  = ((unsigned)(d - 2 + kd) < DIN);        // scalar
            const int offU = (ci * DIN + kd) * (HIN * WIN)
                           + kh * WIN + kwBase;                     // scalar

            // ---- per-lane validity: depth(scalar) & row(vector) & window ----
            const int  ih  = ih0 + kh;
            const bool vdh = vd & ((unsigned)ih < HIN);
            const bool v0  = vdh & (par ? vw0B : vw0A);
            const bool v1  = vdh & (par ? vw1B : vw1A);

            // ---- B fragment: 2 branchless gathers ----
            const int addr0 = laneBase + offU;
            v2f b;
            b.x = gather(x, addr0,     v0);
            b.y = gather(x, addr0 + 1, v1);

            // ---- A fragments from LDS (pair-interleaved: b64 each) ----
            const int abase = (2 * il + hi) * 64 + lo;
            v2f a0 = wlds2[abase];
            v2f a1 = wlds2[abase + 16];
            v2f a2 = wlds2[abase + 32];
            v2f a3 = wlds2[abase + 48];

            acc0 = __builtin_amdgcn_wmma_f32_16x16x4_f32(false, a0, false, b,
                                                         (short)0, acc0, false, false);
            acc1 = __builtin_amdgcn_wmma_f32_16x16x4_f32(false, a1, false, b,
                                                         (short)0, acc1, false, false);
            acc2 = __builtin_amdgcn_wmma_f32_16x16x4_f32(false, a2, false, b,
                                                         (short)0, acc2, false, false);
            acc3 = __builtin_amdgcn_wmma_f32_16x16x4_f32(false, a3, false, b,
                                                         (short)0, acc3, false, false);

            // ---- branchless uniform tap advance (only after odd step) ----
            kh += par;
            const bool chh = (kh == KHN);
            kh = chh ? 0 : kh;
            kd += chh ? 1 : 0;
            const bool cdd = (kd == KDN);
            kd = cdd ? 0 : kd;
            ci += cdd ? 1 : 0;
        }
    }

    // ---- epilogue: bias add + store (coalesced 64B per 16-lane group) ----
    const int outbase = (n * COUT * DOUT + d) * SPATIAL + ptile * 16 + lo;
    #pragma unroll
    for (int r = 0; r < 8; ++r) {
        const int co0 = hi * 8 + r;                 // C layout: M = r + 8*hi
        out[outbase + (co0)      * (DOUT * SPATIAL)] = acc0[r] + bias[co0];
        out[outbase + (co0 + 16) * (DOUT * SPATIAL)] = acc1[r] + bias[co0 + 16];
        out[outbase + (co0 + 32) * (DOUT * SPATIAL)] = acc2[r] + bias[co0 + 32];
        out[outbase + (co0 + 48) * (DOUT * SPATIAL)] = acc3[r] + bias[co0 + 48];
    }
}

extern "C" void kernel_launch(void* const* d_in, const int* in_sizes, int n_in,
                              void* d_out, int out_size, void* d_ws, size_t ws_size,
                              hipStream_t stream) {
    const float* x      = (const float*)d_in[0];
    const float* weight = (const float*)d_in[1];
    const float* bias   = (const float*)d_in[2];
    float* out = (float*)d_out;
    float* wpk = (float*)d_ws;      // 840*64*4 = 215040 bytes of scratch

    const int packN = KTAPS * COUT;                     // 53760
    pack_weights<<<(packN + 255) / 256, 256, 0, stream>>>(weight, wpk);

    const int nTiles  = NBATCH * DOUT * PTILES;         // 25088 wave tiles
    const int nBlocks = nTiles / 8;                     // 3136 blocks, 8 waves each
    conv3d_wmma<<<nBlocks, 256, 0, stream>>>(x, wpk, bias, out);
}